// OldMaskLayer_70016556859456
// MI455X (gfx1250) — compile-verified
//
#include <hip/hip_runtime.h>
#include <stdint.h>

// Problem geometry (fixed by the reference).
#define N_IMG   28
#define HW      784            // 28*28
#define NCH     1024
#define NBATCH  64

// Tiling: one block = (1 batch, 32 channels); 128 threads = 8 channel-groups
// (4 ch each, one float4/B128) x 16 spatial chunks of 49 positions.
// LDS/block ~104.6 KB -> 3 workgroups resident per WGP (320 KB LDS).
#define CB      32
#define NQ      16
#define QLEN    49             // HW / NQ
#define TPB     128
#define NSTAGE  7              // 7 scan stages x 7 positions
#define SLEN    7

typedef float f32x4 __attribute__((ext_vector_type(4)));
typedef int   i32x4 __attribute__((ext_vector_type(4)));

#define AS1 __attribute__((address_space(1)))
#define AS3 __attribute__((address_space(3)))

// ---- CDNA5 async global->LDS DMA (ASYNCcnt path) -------------------------
__device__ __forceinline__ void async_ld_b128(const float* gsrc, float* ldst) {
#if __has_builtin(__builtin_amdgcn_global_load_async_to_lds_b128)
    __builtin_amdgcn_global_load_async_to_lds_b128(
        (AS1 i32x4*)gsrc, (AS3 i32x4*)ldst, /*offset=*/0, /*cpol=*/0);
#else
    unsigned loff = (unsigned)(uint64_t)(AS3 void*)ldst;
    asm volatile("global_load_async_to_lds_b128 %0, %1, off"
                 :: "v"(loff), "v"(gsrc) : "memory");
#endif
}

template <int NREM>
__device__ __forceinline__ void wait_async_le() {
#if __has_builtin(__builtin_amdgcn_s_wait_asynccnt)
    __builtin_amdgcn_s_wait_asynccnt(NREM);
#else
    asm volatile("s_wait_asynccnt %0" :: "i"(NREM) : "memory");
#endif
    asm volatile("" ::: "memory");   // keep LDS reads after the wait
}

// One pipelined scan stage: async loads complete IN ORDER, so once
// asynccnt <= 49 - 7*(C+1), positions [7C, 7C+7) are guaranteed in LDS.
template <int C>
__device__ __forceinline__ void scan_stage(const float* sval, int p0, int cg,
                                           f32x4& vmax, i32x4& vpos) {
    wait_async_le<QLEN - SLEN * (C + 1)>();
#pragma unroll
    for (int k = SLEN * C; k < SLEN * (C + 1); ++k) {
        const int p = p0 + k;
        f32x4 v = *(const f32x4*)(sval + p * CB + cg);
#pragma unroll
        for (int j = 0; j < 4; ++j) {
            if (v[j] > vmax[j]) { vmax[j] = v[j]; vpos[j] = p; }  // first-occurrence
        }
    }
}

__global__ __launch_bounds__(TPB)
void peak_mask_kernel(const float* __restrict__ in, float* __restrict__ out) {
    extern __shared__ float smem[];
    float* sval = smem;                        // [HW][CB]  values (~98 KB)
    float* pmax = smem + HW * CB;              // [NQ][CB]  partial maxima
    int*   pidx = (int*)(pmax + NQ * CB);      // [NQ][CB]  partial arg indices
    int*   fij  = (int*)(pidx + NQ * CB);      // [CB]      packed (i<<8)|j

    const int tid = threadIdx.x;
    const int g   = tid & 7;                   // channel group (4 channels)
    const int q   = tid >> 3;                  // spatial chunk (0..15)
    const int b   = blockIdx.x >> 5;           // batch
    const int cb  = (blockIdx.x & 31) * CB;    // channel base of this block
    const int cg  = 4 * g;                     // channel offset inside block
    const int p0  = q * QLEN;

    const float* gbase = in + (size_t)b * HW * NCH + cb + cg;

    // ---- Stage in: 49 async B128 DMAs per thread, HBM -> LDS (single HBM read)
    for (int k = 0; k < QLEN; ++k) {
        const int p = p0 + k;
        async_ld_b128(gbase + (size_t)p * NCH, sval + p * CB + cg);
    }

    // ---- Pass 1: per-channel argmax, pipelined against the in-flight DMAs.
    f32x4 vmax = { -INFINITY, -INFINITY, -INFINITY, -INFINITY };
    i32x4 vpos = { 0, 0, 0, 0 };
    scan_stage<0>(sval, p0, cg, vmax, vpos);
    scan_stage<1>(sval, p0, cg, vmax, vpos);
    scan_stage<2>(sval, p0, cg, vmax, vpos);
    scan_stage<3>(sval, p0, cg, vmax, vpos);
    scan_stage<4>(sval, p0, cg, vmax, vpos);
    scan_stage<5>(sval, p0, cg, vmax, vpos);
    scan_stage<6>(sval, p0, cg, vmax, vpos);   // final: asynccnt <= 0

    *(f32x4*)(pmax + q * CB + cg) = vmax;
    *(i32x4*)(pidx + q * CB + cg) = vpos;
    __syncthreads();

    // ---- Cross-chunk reduce (ascending q + strict '>' == jnp.argmax ties).
    if (tid < CB) {
        float best = pmax[tid];
        int   bi   = pidx[tid];
#pragma unroll
        for (int qq = 1; qq < NQ; ++qq) {
            const float v = pmax[qq * CB + tid];
            const int   i = pidx[qq * CB + tid];
            if (v > best) { best = v; bi = i; }
        }
        const int im = bi / N_IMG;
        const int jm = bi - im * N_IMG;
        fij[tid] = (im << 8) | jm;
    }
    __syncthreads();

    // ---- Pass 2: triangular peak mask from LDS-resident data, store out.
    const i32x4 f4 = *(const i32x4*)(fij + cg);
    int im[4], jm[4];
#pragma unroll
    for (int j = 0; j < 4; ++j) { im[j] = f4[j] >> 8; jm[j] = f4[j] & 255; }

    const float tau = 0.5f / (float)HW;
    float* obase = out + (size_t)b * HW * NCH + cb + cg;
    int pi = p0 / N_IMG;
    int pj = p0 - pi * N_IMG;
    for (int k = 0; k < QLEN; ++k) {
        const int p = p0 + k;
        f32x4 v = *(const f32x4*)(sval + p * CB + cg);
        f32x4 r;
#pragma unroll
        for (int j = 0; j < 4; ++j) {
            int di = pi - im[j]; di = di < 0 ? -di : di;
            int dj = pj - jm[j]; dj = dj < 0 ? -dj : dj;
            const float d = (float)(di + dj);
            const float t = 1.0f - (4.0f * d) * (1.0f / 28.0f);   // beta*dist/n
            r[j] = v[j] * (tau * fmaxf(t, -1.0f));
        }
        *(f32x4*)(obase + (size_t)p * NCH) = r;
        if (++pj == N_IMG) { pj = 0; ++pi; }
    }
}

extern "C" void kernel_launch(void* const* d_in, const int* in_sizes, int n_in,
                              void* d_out, int out_size, void* d_ws, size_t ws_size,
                              hipStream_t stream) {
    const float* in  = (const float*)d_in[0];
    float*       out = (float*)d_out;

    // LDS: values 98 KB + partials 4 KB + peak coords 128 B = 104,576 B
    // -> 3 concurrent workgroups per WGP (3 * 104,576 = 313,728 < 327,680).
    const size_t lds_bytes = (size_t)(HW * CB + NQ * CB) * sizeof(float)
                           + (size_t)(NQ * CB + CB) * sizeof(int);

    dim3 grid(NBATCH * (NCH / CB));   // 64 batches * 32 channel-blocks = 2048
    peak_mask_kernel<<<grid, dim3(TPB), lds_bytes, stream>>>(in, out);
}